// gae_69827578298829
// MI455X (gfx1250) — compile-verified
//
#include <hip/hip_runtime.h>

typedef __attribute__((ext_vector_type(2))) float v2f;
typedef __attribute__((ext_vector_type(8))) float v8f;

// ---------------- zero init (accumulation buffers) ----------------
__global__ void zero_f32(float* __restrict__ p, long n) {
    long i = (long)blockIdx.x * blockDim.x + threadIdx.x;
    long stride = (long)gridDim.x * blockDim.x;
    for (; i < n; i += stride) p[i] = 0.0f;
}

// ---------------- edge scatter-add: agg[dst] += feat[src] ----------------
template <int F, int LOGF>
__global__ void edge_scatter(const float* __restrict__ feat,
                             const int* __restrict__ src,
                             const int* __restrict__ dst,
                             float* __restrict__ agg, long total) {
    long i = (long)blockIdx.x * blockDim.x + threadIdx.x;
    if (i >= total) return;
    long e = i >> LOGF;
    int  f = (int)(i & (F - 1));
    int  s = src[e];
    int  d = dst[e];
    atomicAdd(&agg[(long)d * F + f], feat[(long)s * F + f]);
}

// ---------------- fused GraphConv GEMM (fp32 WMMA) ----------------
// out[M,NOUT] = relu(Aagg @ Wrel^T + Aroot @ Wroot^T + bias)
// Wrel/Wroot stored row-major [NOUT, K] (so B(k,n) = W[n*K + k]).
// One wave32 computes one 16x16 output tile using V_WMMA_F32_16X16X4_F32.
template <int K, int NOUT>
__global__ void graphconv_gemm(const float* __restrict__ Aagg,
                               const float* __restrict__ Aroot,
                               const float* __restrict__ Wrel,
                               const float* __restrict__ Wroot,
                               const float* __restrict__ bias,
                               float* __restrict__ out, int M) {
    const int lane = threadIdx.x & 31;
    const int wave = threadIdx.x >> 5;
    const int NT = NOUT / 16;
    const int tile = blockIdx.x * (blockDim.x >> 5) + wave;
    const int totalTiles = (M / 16) * NT;
    if (tile >= totalTiles) return;          // wave-uniform: EXEC stays all-1s
    const int mt = tile / NT;
    const int nt = tile % NT;

    // fp32 WMMA A layout: lanes 0-15 hold row M=lane, K = kb+{0,1};
    //                     lanes 16-31 hold row M=lane-16, K = kb+{2,3}.
    // B layout mirrors with N=lane&15 (column of W^T = row of W).
    const int row = mt * 16 + (lane & 15);
    const int col = nt * 16 + (lane & 15);
    const int kl  = (lane >> 4) * 2;         // 0 or 2

    v8f acc = {};
    const float* arow = Aagg  + (long)row * K;
    const float* brow = Wrel  + (long)col * K;
#pragma unroll
    for (int kb = 0; kb < K; kb += 4) {
        v2f a = *(const v2f*)(arow + kb + kl);   // 8-byte aligned: kb+kl even
        v2f b = *(const v2f*)(brow + kb + kl);
        acc = __builtin_amdgcn_wmma_f32_16x16x4_f32(
            false, a, false, b, (short)0, acc, false, false);
    }
    arow = Aroot + (long)row * K;
    brow = Wroot + (long)col * K;
#pragma unroll
    for (int kb = 0; kb < K; kb += 4) {
        v2f a = *(const v2f*)(arow + kb + kl);
        v2f b = *(const v2f*)(brow + kb + kl);
        acc = __builtin_amdgcn_wmma_f32_16x16x4_f32(
            false, a, false, b, (short)0, acc, false, false);
    }

    // C/D layout: VGPR v -> row M = v + 8*(lane>>4), col N = lane&15
    const float bv = bias[col];
    const int rbase = mt * 16 + (lane >> 4) * 8;
#pragma unroll
    for (int v = 0; v < 8; ++v) {
        float r = acc[v] + bv;
        out[(long)(rbase + v) * NOUT + col] = r > 0.0f ? r : 0.0f;
    }
}

// ---------------- global mean pool ----------------
__global__ void pool_accum(const float* __restrict__ h2,
                           const int* __restrict__ batch,
                           float* __restrict__ pool, long total) {
    long i = (long)blockIdx.x * blockDim.x + threadIdx.x;
    if (i >= total) return;
    long n = i >> 6;
    int  f = (int)(i & 63);
    atomicAdd(&pool[(long)batch[n] * 64 + f], h2[i]);
}

__global__ void pool_count(const int* __restrict__ batch,
                           float* __restrict__ cnt, int Nn) {
    int n = blockIdx.x * blockDim.x + threadIdx.x;
    if (n < Nn) atomicAdd(&cnt[batch[n]], 1.0f);
}

__global__ void pool_final(const float* __restrict__ pool,
                           const float* __restrict__ cnt,
                           float* __restrict__ h3out) {
    int i = blockIdx.x * blockDim.x + threadIdx.x;   // 64*64 threads
    int g = i >> 6;
    h3out[i] = pool[i] / fmaxf(cnt[g], 1.0f);
}

// ---------------- encoder/decoder MLPs (G=64 rows, trivially small) ----------------
__global__ void mlp_kernel(const float* __restrict__ h3,
                           const float* __restrict__ es_w1, const float* __restrict__ es_b1,
                           const float* __restrict__ es_w2, const float* __restrict__ es_b2,
                           const float* __restrict__ es_w3, const float* __restrict__ es_b3,
                           const float* __restrict__ d_w1,  const float* __restrict__ d_b1,
                           const float* __restrict__ d_w2,  const float* __restrict__ d_b2,
                           const float* __restrict__ d_w3,  const float* __restrict__ d_b3,
                           float* __restrict__ xout) {
    int g = threadIdx.x;   // 64 threads, one graph each
    float h[64];
#pragma unroll
    for (int c = 0; c < 64; ++c) h[c] = h3[g * 64 + c];
    float z1[8];
#pragma unroll
    for (int o = 0; o < 8; ++o) {
        float s = es_b1[o];
        for (int c = 0; c < 64; ++c) s += es_w1[o * 64 + c] * h[c];
        z1[o] = fmaxf(s, 0.0f);
    }
    float z2[4];
#pragma unroll
    for (int o = 0; o < 4; ++o) {
        float s = es_b2[o];
        for (int c = 0; c < 8; ++c) s += es_w2[o * 8 + c] * z1[c];
        z2[o] = fmaxf(s, 0.0f);
    }
    float z = es_b3[0];
#pragma unroll
    for (int c = 0; c < 4; ++c) z += es_w3[c] * z2[c];
    float t1[4];
#pragma unroll
    for (int o = 0; o < 4; ++o) t1[o] = fmaxf(d_b1[o] + d_w1[o] * z, 0.0f);
    float t2[8];
#pragma unroll
    for (int o = 0; o < 8; ++o) {
        float s = d_b2[o];
        for (int c = 0; c < 4; ++c) s += d_w2[o * 4 + c] * t1[c];
        t2[o] = fmaxf(s, 0.0f);
    }
#pragma unroll
    for (int o = 0; o < 64; ++o) {
        float s = d_b3[o];
        for (int c = 0; c < 8; ++c) s += d_w3[o * 8 + c] * t2[c];
        xout[g * 64 + o] = s;
    }
}

extern "C" void kernel_launch(void* const* d_in, const int* in_sizes, int n_in,
                              void* d_out, int out_size, void* d_ws, size_t ws_size,
                              hipStream_t stream) {
    const float* x       = (const float*)d_in[0];
    const int*   eidx    = (const int*)  d_in[1];
    const int*   batch   = (const int*)  d_in[2];
    const float* w1_rel  = (const float*)d_in[3];
    const float* b1_rel  = (const float*)d_in[4];
    const float* w1_root = (const float*)d_in[5];
    const float* w2_rel  = (const float*)d_in[6];
    const float* b2_rel  = (const float*)d_in[7];
    const float* w2_root = (const float*)d_in[8];
    const float* es_w1   = (const float*)d_in[9];
    const float* es_b1   = (const float*)d_in[10];
    const float* es_w2   = (const float*)d_in[11];
    const float* es_b2   = (const float*)d_in[12];
    const float* es_w3   = (const float*)d_in[13];
    const float* es_b3   = (const float*)d_in[14];
    const float* d_w1    = (const float*)d_in[15];
    const float* d_b1    = (const float*)d_in[16];
    const float* d_w2    = (const float*)d_in[17];
    const float* d_b2    = (const float*)d_in[18];
    const float* d_w3    = (const float*)d_in[19];
    const float* d_b3    = (const float*)d_in[20];

    const int Nn = in_sizes[0] / 64;       // 100000
    const int E  = in_sizes[1] / 2;        // 1250000
    const int* src = eidx;
    const int* dst = eidx + E;

    // workspace layout (floats): accumulated buffers first (zeroed each call)
    float* ws   = (float*)d_ws;
    float* agg1 = ws;                                  // Nn*64
    float* agg2 = agg1 + (size_t)Nn * 64;              // Nn*128
    float* pool = agg2 + (size_t)Nn * 128;             // 64*64
    float* cnt  = pool + 64 * 64;                      // 64
    float* h1   = cnt + 64;                            // Nn*128
    float* h2   = h1 + (size_t)Nn * 128;               // Nn*64

    float* h3_out = (float*)d_out;                     // [64,64]
    float* x_out  = (float*)d_out + 64 * 64;           // [64,64]

    const long zeroN = (long)Nn * 64 + (long)Nn * 128 + 64 * 64 + 64;
    zero_f32<<<2048, 256, 0, stream>>>(ws, zeroN);

    // Layer 1: scatter + fused GEMM (K=64 -> NOUT=128)
    {
        long total = (long)E * 64;
        edge_scatter<64, 6><<<(int)((total + 255) / 256), 256, 0, stream>>>(x, src, dst, agg1, total);
        int tiles = (Nn / 16) * (128 / 16);
        graphconv_gemm<64, 128><<<(tiles + 7) / 8, 256, 0, stream>>>(
            agg1, x, w1_rel, w1_root, b1_rel, h1, Nn);
    }
    // Layer 2: scatter + fused GEMM (K=128 -> NOUT=64)
    {
        long total = (long)E * 128;
        edge_scatter<128, 7><<<(int)((total + 255) / 256), 256, 0, stream>>>(h1, src, dst, agg2, total);
        int tiles = (Nn / 16) * (64 / 16);
        graphconv_gemm<128, 64><<<(tiles + 7) / 8, 256, 0, stream>>>(
            agg2, h1, w2_rel, w2_root, b2_rel, h2, Nn);
    }
    // Mean pool -> h3 (output 0)
    {
        long total = (long)Nn * 64;
        pool_accum<<<(int)((total + 255) / 256), 256, 0, stream>>>(h2, batch, pool, total);
        pool_count<<<(Nn + 255) / 256, 256, 0, stream>>>(batch, cnt, Nn);
        pool_final<<<16, 256, 0, stream>>>(pool, cnt, h3_out);
    }
    // Encoder/decoder MLPs -> x_ (output 1)
    mlp_kernel<<<1, 64, 0, stream>>>(h3_out,
                                     es_w1, es_b1, es_w2, es_b2, es_w3, es_b3,
                                     d_w1, d_b1, d_w2, d_b2, d_w3, d_b3,
                                     x_out);
}